// PartitionsToFeatures_68023692034126
// MI455X (gfx1250) — compile-verified
//
#include <hip/hip_runtime.h>
#include <hip/hip_bf16.h>
#include <stdint.h>

#ifndef __has_builtin
#define __has_builtin(x) 0
#endif

#define AS1 __attribute__((address_space(1)))
#define AS3 __attribute__((address_space(3)))

typedef int v4i __attribute__((ext_vector_type(4)));   // matches builtin's V4i pointee

// gfx12.5 CPol: TH in bits [2:0]; TH=0 RT (default), TH=1 NT (non-temporal)
#define CPOL_RT      0
#define CPOL_ST_NT   1

// ---- CDNA5 async global<->LDS DMA helpers (ASYNCcnt-tracked, no VGPR data path) ----

__device__ __forceinline__ void async_g2l_b128(const void* gsrc, void* ldst) {
#if __has_builtin(__builtin_amdgcn_global_load_async_to_lds_b128)
    // RT hint: input is re-read ~4x by overlapping shifted windows; 134MB fits in 192MB L2
    __builtin_amdgcn_global_load_async_to_lds_b128((AS1 v4i*)gsrc, (AS3 v4i*)ldst,
                                                   /*offset=*/0, /*cpol=*/CPOL_RT);
#else
    unsigned laddr = (unsigned)(unsigned long long)(AS3 char*)ldst;
    asm volatile("global_load_async_to_lds_b128 %0, %1, off"
                 :: "v"(laddr), "v"((unsigned long long)gsrc) : "memory");
#endif
}

__device__ __forceinline__ void async_l2g_b128(void* gdst, const void* lsrc) {
#if __has_builtin(__builtin_amdgcn_global_store_async_from_lds_b128)
    // NT hint: 504MB output is write-once/never-read -> don't displace input from L2
    __builtin_amdgcn_global_store_async_from_lds_b128((AS1 v4i*)gdst, (AS3 v4i*)lsrc,
                                                      /*offset=*/0, /*cpol=*/CPOL_ST_NT);
#else
    unsigned laddr = (unsigned)(unsigned long long)(AS3 char*)lsrc;
    asm volatile("global_store_async_from_lds_b128 %0, %1, off th:TH_STORE_NT"
                 :: "v"((unsigned long long)gdst), "v"(laddr) : "memory");
#endif
}

__device__ __forceinline__ void wait_asynccnt0() {
#if __has_builtin(__builtin_amdgcn_s_wait_asynccnt)
    __builtin_amdgcn_s_wait_asynccnt(0);
#else
    asm volatile("s_wait_asynccnt 0x0" ::: "memory");
#endif
}

// ---------------------------------------------------------------------------------
// x: (8, 256, 256, 64) f32, WIN=16, shifts (0,0),(0,8),(8,0),(8,8)
// out: (8, 961, 16384) f32, windows concatenated in shift order.
//
// One block = one 4KB contiguous chunk: (b, window p, window-row wr) -> the 16*64
// floats of that window row, contiguous in BOTH input and output.
// chunk = ((b*961 + p)*16 + wr);  dst = out + chunk*1024
// src = x + ((b*256 + row)*256 + col0)*64, row = r0 + wi*16 + wr, col0 = c0 + wj*16
// ---------------------------------------------------------------------------------

__global__ __launch_bounds__(256)
void PartitionsToFeatures_68023692034126_kernel(const float* __restrict__ x,
                                                float* __restrict__ out) {
    constexpr int H  = 256;
    constexpr int W  = 256;
    constexpr int C  = 64;
    constexpr int NP = 961;               // 256 + 240 + 240 + 225 windows per batch

    __shared__ float4 buf[256];           // 4KB staging tile

    const int t     = threadIdx.x;
    const int chunk = blockIdx.x;         // uniform -> SALU address math
    const int wr    = chunk & 15;         // window-row
    const int bp    = chunk >> 4;
    const int b     = bp / NP;            // constant divisor -> magic number
    const int p     = bp - b * NP;

    // Decompose concatenated window index p into (shift k, wi, wj).
    // Segment sizes: k0: 16x16=256, k1: 16x15=240, k2: 15x16=240, k3: 15x15=225
    int row, col0;
    if (p < 256) {                         // r0=0, c0=0, nc=16
        int wi = p >> 4, wj = p & 15;
        row  = wi * 16 + wr;
        col0 = wj * 16;
    } else if (p < 496) {                  // r0=0, c0=8, nc=15
        int pl = p - 256;
        int wi = pl / 15, wj = pl - wi * 15;
        row  = wi * 16 + wr;
        col0 = 8 + wj * 16;
    } else if (p < 736) {                  // r0=8, c0=0, nc=16
        int pl = p - 496;
        int wi = pl >> 4, wj = pl & 15;
        row  = 8 + wi * 16 + wr;
        col0 = wj * 16;
    } else {                               // r0=8, c0=8, nc=15
        int pl = p - 736;
        int wi = pl / 15, wj = pl - wi * 15;
        row  = 8 + wi * 16 + wr;
        col0 = 8 + wj * 16;
    }

    const float* src = x   + ((size_t)(b * H + row) * W + col0) * C; // 1024 floats, contig
    float*       dst = out + (size_t)chunk * 1024;                   // 1024 floats, contig

    // Stream 16B per lane: global -> LDS (async DMA), drain, LDS -> global (async DMA).
    // Lane t loads and stores buf[t] only => wave-private, no barrier required.
    async_g2l_b128(src + 4 * t, &buf[t]);
    wait_asynccnt0();                      // LDS data visible before async store reads it
    async_l2g_b128(dst + 4 * t, &buf[t]);
    // trailing async stores are drained by s_endpgm's implicit wait-idle
}

extern "C" void kernel_launch(void* const* d_in, const int* in_sizes, int n_in,
                              void* d_out, int out_size, void* d_ws, size_t ws_size,
                              hipStream_t stream) {
    (void)in_sizes; (void)n_in; (void)out_size; (void)d_ws; (void)ws_size;
    const float* x   = (const float*)d_in[0];
    float*       out = (float*)d_out;

    // 8 batches * 961 windows * 16 window-rows = 123,008 chunks of 4KB
    const int chunks = 8 * 961 * 16;
    PartitionsToFeatures_68023692034126_kernel<<<dim3(chunks), dim3(256), 0, stream>>>(x, out);
}